// StudentNet_22943715295509
// MI455X (gfx1250) — compile-verified
//
#include <hip/hip_runtime.h>
#include <hip/hip_bf16.h>
#include <math.h>

typedef float v2f __attribute__((ext_vector_type(2)));
typedef float v8f __attribute__((ext_vector_type(8)));

// ---------------------------------------------------------------------------
// pack [Wg | Wl | 0-pad] into K-pair-interleaved layout:
//   Wc[((k>>1)*NC + j)*2 + (k&1)] = w[k][j]
// so a WMMA B-fragment (two consecutive K values at one column) is one b64 load
// ---------------------------------------------------------------------------
__global__ void pack_w_pairs(const float* __restrict__ Wg, const float* __restrict__ Wl,
                             float* __restrict__ Wc, int K, int gc, int lc, int NC) {
    int i = blockIdx.x * blockDim.x + threadIdx.x;
    if (i >= K * NC) return;
    int k = i / NC, j = i % NC;
    float v = 0.0f;
    if (j < gc)           v = Wg[k * gc + j];
    else if (j < gc + lc) v = Wl[k * lc + (j - gc)];
    Wc[((size_t)(k >> 1) * NC + j) * 2 + (k & 1)] = v;
}

// ---------------------------------------------------------------------------
// f32 WMMA GEMM:  Y[M x NC] = X[M x K] @ W[K x NC]  (W in pair-interleaved form)
// one wave per 16-row stripe, NT 16-col tiles in v8f accumulators,
// K stepped by 4 with v_wmma_f32_16x16x4_f32 (full f32 precision).
// A frag (ISA 7.12.2): lane = half*16 + m ; VGPR i -> K = half*2 + i.
// Out-of-range rows read clamped row 0 (harmless: their D rows are not stored).
// ---------------------------------------------------------------------------
template <int NT>
__global__ __launch_bounds__(128) void gemm_wmma_f32(
    const float* __restrict__ X, const float* __restrict__ W,
    float* __restrict__ Y, int M, int K) {
    constexpr int NC = NT * 16;
    const int lane = threadIdx.x & 31;
    const int wave = threadIdx.x >> 5;
    const int row0 = (blockIdx.x * 4 + wave) * 16;
    if (row0 >= M) return;                       // wave-uniform exit, EXEC stays full
    const int mi = lane & 15;                    // M index (A) / N index (B)
    const int kh = (lane >> 4) << 1;             // 0 or 2 (K sub-offset per lane half)
    const int row = row0 + mi;
    const float* xr = X + (size_t)(row < M ? row : 0) * K;

    v8f zero = {0.f, 0.f, 0.f, 0.f, 0.f, 0.f, 0.f, 0.f};
    v8f acc[NT];
#pragma unroll
    for (int t = 0; t < NT; ++t) acc[t] = zero;

    for (int k = 0; k < K; k += 4) {             // K is 128 or 136: multiple of 4
        const int k0 = k + kh;
        v2f a = *(const v2f*)(xr + k0);                      // one b64 load
        const float* wr = W + (size_t)(k0 >> 1) * NC * 2;
#pragma unroll
        for (int t = 0; t < NT; ++t) {
            v2f b = *(const v2f*)(wr + (t * 16 + mi) * 2);   // one b64 load
            acc[t] = __builtin_amdgcn_wmma_f32_16x16x4_f32(
                false, a, false, b, (short)0, acc[t], false, false);
        }
    }
    // C/D layout: VGPR i -> M = i + (lane>=16 ? 8 : 0), N = lane & 15
    const int rb = row0 + ((lane >> 4) ? 8 : 0);
#pragma unroll
    for (int t = 0; t < NT; ++t) {
#pragma unroll
        for (int i = 0; i < 8; ++i) {
            const int r = rb + i;
            if (r < M) Y[(size_t)r * NC + t * 16 + mi] = acc[t][i];
        }
    }
}

// ---------------------------------------------------------------------------
// per-(node,head) attention scalars: a_src[n,h] = <h[n,h,:], ag[0,h,:]>, etc.
// h lives in the first columns of Y (row stride ldY). ag shape is [2, 2, C].
// ---------------------------------------------------------------------------
__global__ void attn_vec(const float* __restrict__ Y, const float* __restrict__ ag,
                         float* __restrict__ asrc, float* __restrict__ adst,
                         int n, int ldY, int C) {
    int i = blockIdx.x * blockDim.x + threadIdx.x;   // node*2 + head
    if (i >= n * 2) return;
    int node = i >> 1, h = i & 1;
    const float* hv = Y + (size_t)node * ldY + h * C;
    const float* a0 = ag + h * C;            // ag[0][h]
    const float* a1 = ag + 2 * C + h * C;    // ag[1][h]
    float s0 = 0.f, s1 = 0.f;
    for (int c = 0; c < C; ++c) {
        float v = hv[c];
        s0 += v * a0[c];
        s1 += v * a1[c];
    }
    asrc[i] = s0;
    adst[i] = s1;
}

// ---------------------------------------------------------------------------
// CSR construction (edge list is a constant input -> build once, reuse 5x)
// ---------------------------------------------------------------------------
__global__ void fill_i32(int* __restrict__ p, int v, int count) {
    int i = blockIdx.x * blockDim.x + threadIdx.x;
    if (i < count) p[i] = v;
}

__global__ void count_deg(const long long* __restrict__ dstI, int* __restrict__ deg,
                          int nedges) {
    int e = blockIdx.x * blockDim.x + threadIdx.x;
    if (e >= nedges) return;
    atomicAdd(&deg[(int)dstI[e]], 1);
}

// single-workgroup exclusive scan over deg[0..n) -> rowptr (and cursor copy)
__global__ __launch_bounds__(1024) void build_rowptr(
    const int* __restrict__ deg, int* __restrict__ rowptr,
    int* __restrict__ cursor, int n) {
    __shared__ int part[1024];
    const int t = threadIdx.x;
    const int chunk = (n + 1023) / 1024;
    const int beg = t * chunk;
    const int end = (beg + chunk < n) ? beg + chunk : n;
    int s = 0;
    for (int i = beg; i < end; ++i) s += deg[i];
    part[t] = s;
    __syncthreads();
    for (int off = 1; off < 1024; off <<= 1) {      // Hillis-Steele inclusive scan
        int v = part[t];
        int u = (t >= off) ? part[t - off] : 0;
        __syncthreads();
        part[t] = v + u;
        __syncthreads();
    }
    int excl = (t == 0) ? 0 : part[t - 1];
    for (int i = beg; i < end; ++i) {
        rowptr[i] = excl;
        cursor[i] = excl;
        excl += deg[i];
    }
    if (t == 1023) rowptr[n] = part[1023];
}

__global__ void fill_edges(const long long* __restrict__ srcI,
                           const long long* __restrict__ dstI,
                           int* __restrict__ cursor, int* __restrict__ srclist,
                           int nedges) {
    int e = blockIdx.x * blockDim.x + threadIdx.x;
    if (e >= nedges) return;
    int d = (int)dstI[e];
    int pos = atomicAdd(&cursor[d], 1);
    srclist[pos] = (int)srcI[e];
}

// ---------------------------------------------------------------------------
// fused segment-softmax + aggregation, one wave per destination node.
// pass1/2: lane-parallel over incident edges, shfl_xor butterflies.
// pass3: edges serial, channels lane-parallel; direct stores -> ZERO atomics.
// ---------------------------------------------------------------------------
__global__ __launch_bounds__(256) void gat_aggregate_csr(
    const int* __restrict__ rowptr, const int* __restrict__ srclist,
    const float* __restrict__ asrc, const float* __restrict__ adst,
    const float* __restrict__ Y, float* __restrict__ agg,
    int n, int ldY, int HC, int C) {
    const int d = blockIdx.x * (blockDim.x >> 5) + (threadIdx.x >> 5);
    const int lane = threadIdx.x & 31;
    if (d >= n) return;
    const int beg = rowptr[d], end = rowptr[d + 1];
    const float ad0 = adst[d * 2 + 0], ad1 = adst[d * 2 + 1];

    // pass 1: per-head segment max
    float m0 = -INFINITY, m1 = -INFINITY;
    for (int j = beg + lane; j < end; j += 32) {
        int s = srclist[j];
        float e0 = asrc[s * 2 + 0] + ad0; e0 = e0 > 0.f ? e0 : 0.2f * e0;
        float e1 = asrc[s * 2 + 1] + ad1; e1 = e1 > 0.f ? e1 : 0.2f * e1;
        m0 = fmaxf(m0, e0); m1 = fmaxf(m1, e1);
    }
#pragma unroll
    for (int off = 16; off; off >>= 1) {
        m0 = fmaxf(m0, __shfl_xor(m0, off, 32));
        m1 = fmaxf(m1, __shfl_xor(m1, off, 32));
    }

    // pass 2: per-head softmax denominator
    float s0 = 0.f, s1 = 0.f;
    for (int j = beg + lane; j < end; j += 32) {
        int s = srclist[j];
        float e0 = asrc[s * 2 + 0] + ad0; e0 = e0 > 0.f ? e0 : 0.2f * e0;
        float e1 = asrc[s * 2 + 1] + ad1; e1 = e1 > 0.f ? e1 : 0.2f * e1;
        s0 += expf(e0 - m0); s1 += expf(e1 - m1);
    }
#pragma unroll
    for (int off = 16; off; off >>= 1) {
        s0 += __shfl_xor(s0, off, 32);
        s1 += __shfl_xor(s1, off, 32);
    }
    const float inv0 = 1.f / (s0 + 1e-16f), inv1 = 1.f / (s1 + 1e-16f);

    // pass 3: alpha-weighted gather of h[src]
    float acc[5];                                    // ceil(136/32) channel chunks
#pragma unroll
    for (int cc = 0; cc < 5; ++cc) acc[cc] = 0.f;
    for (int j = beg; j < end; ++j) {
        int s = srclist[j];
        float e0 = asrc[s * 2 + 0] + ad0; e0 = e0 > 0.f ? e0 : 0.2f * e0;
        float e1 = asrc[s * 2 + 1] + ad1; e1 = e1 > 0.f ? e1 : 0.2f * e1;
        float w0 = expf(e0 - m0) * inv0;
        float w1 = expf(e1 - m1) * inv1;
        const float* hs = Y + (size_t)s * ldY;
#pragma unroll
        for (int cc = 0; cc < 5; ++cc) {
            int c = lane + cc * 32;
            if (c < HC) acc[cc] += ((c < C) ? w0 : w1) * hs[c];
        }
    }
    float* ao = agg + (size_t)d * HC;
#pragma unroll
    for (int cc = 0; cc < 5; ++cc) {
        int c = lane + cc * 32;
        if (c < HC) ao[c] = acc[cc];
    }
}

// layers 1-4: x_next = elu(agg + bg + skip + bl)   (in-place over agg buffer)
__global__ void finalize_elu(const float* __restrict__ agg, const float* __restrict__ Y,
                             const float* __restrict__ bg, const float* __restrict__ bl,
                             float* __restrict__ xout, int n, int ldY, int HC) {
    long long i = (long long)blockIdx.x * blockDim.x + threadIdx.x;
    if (i >= (long long)n * HC) return;
    int node = (int)(i / HC), j = (int)(i % HC);
    float v = agg[i] + bg[j] + Y[(size_t)node * ldY + HC + j] + bl[j];
    xout[i] = v > 0.0f ? v : (expf(v) - 1.0f);
}

// layer 5: out = mean_heads(agg) + bg + skip + bl  (no elu)
__global__ void finalize_mean(const float* __restrict__ agg, const float* __restrict__ Y,
                              const float* __restrict__ bg, const float* __restrict__ bl,
                              float* __restrict__ out, int n, int ldY, int C) {
    long long i = (long long)blockIdx.x * blockDim.x + threadIdx.x;
    if (i >= (long long)n * C) return;
    int node = (int)(i / C), j = (int)(i % C);
    int HC = 2 * C;
    float m = 0.5f * (agg[(size_t)node * HC + j] + agg[(size_t)node * HC + C + j]);
    out[i] = m + bg[j] + Y[(size_t)node * ldY + HC + j] + bl[j];
}

// ---------------------------------------------------------------------------
extern "C" void kernel_launch(void* const* d_in, const int* in_sizes, int n_in,
                              void* d_out, int out_size, void* d_ws, size_t ws_size,
                              hipStream_t stream) {
    const int N = 50000, EE = 850000;
    const float* x_in = (const float*)d_in[0];
    const long long* ei = (const long long*)d_in[1];
    const long long* srcI = ei;
    const long long* dstI = ei + EE;

    float* ws    = (float*)d_ws;
    float* cur   = ws;                          // N*136  (layer x, reused as agg out)
    float* Y     = cur + (size_t)N * 136;       // N*272  ([h | skip] GEMM output)
    float* asrc  = Y + (size_t)N * 272;         // N*2
    float* adst  = asrc + (size_t)N * 2;        // N*2
    float* wcat  = adst + (size_t)N * 2;        // 136*272 packed weights
    int* ibuf    = (int*)(wcat + (size_t)136 * 272);
    int* deg     = ibuf;                        // N
    int* rowptr  = deg + N;                     // N+1
    int* cursor  = rowptr + N + 1;              // N
    int* srclist = cursor + N;                  // EE

    // ---- CSR build: once, reused by all 5 layers ----
    fill_i32<<<(N + 255) / 256, 256, 0, stream>>>(deg, 0, N);
    count_deg<<<(EE + 255) / 256, 256, 0, stream>>>(dstI, deg, EE);
    build_rowptr<<<1, 1024, 0, stream>>>(deg, rowptr, cursor, N);
    fill_edges<<<(EE + 255) / 256, 256, 0, stream>>>(srcI, dstI, cursor, srclist, EE);

    for (int L = 0; L < 5; ++L) {
        const float* Wg = (const float*)d_in[2 + 5 * L];
        const float* ag = (const float*)d_in[3 + 5 * L];
        const float* bg = (const float*)d_in[4 + 5 * L];
        const float* Wl = (const float*)d_in[5 + 5 * L];
        const float* bl = (const float*)d_in[6 + 5 * L];
        const int K  = (L == 0) ? 128 : 136;
        const int gc = (L == 4) ? 80 : 136;     // GAT lin output cols (= H*C)
        const int lc = (L == 4) ? 40 : 136;     // skip lin output cols
        const int C  = (L == 4) ? 40 : 68;      // per-head channels
        const int HC = gc;
        const int NC = (L == 4) ? 128 : 272;    // padded GEMM width (NT*16)
        const float* X = (L == 0) ? x_in : cur;

        pack_w_pairs<<<(K * NC + 255) / 256, 256, 0, stream>>>(Wg, Wl, wcat, K, gc, lc, NC);

        int gblocks = (N + 63) / 64;
        if (L == 4) gemm_wmma_f32<8><<<gblocks, 128, 0, stream>>>(X, wcat, Y, N, K);
        else        gemm_wmma_f32<17><<<gblocks, 128, 0, stream>>>(X, wcat, Y, N, K);

        attn_vec<<<(N * 2 + 255) / 256, 256, 0, stream>>>(Y, ag, asrc, adst, N, NC, C);

        gat_aggregate_csr<<<(N + 7) / 8, 256, 0, stream>>>(
            rowptr, srclist, asrc, adst, Y, cur, N, NC, HC, C);

        if (L < 4) {
            long long nfin = (long long)N * HC;
            finalize_elu<<<(unsigned)((nfin + 255) / 256), 256, 0, stream>>>(
                cur, Y, bg, bl, cur, N, NC, HC);
        } else {
            long long nfin = (long long)N * C;
            finalize_mean<<<(unsigned)((nfin + 255) / 256), 256, 0, stream>>>(
                cur, Y, bg, bl, (float*)d_out, N, NC, C);
        }
    }
}